// unit_ctrgcn_36129264894593
// MI455X (gfx1250) — compile-verified
//
#include <hip/hip_runtime.h>
#include <math.h>

#define NN 64
#define CC 64
#define TT 256
#define VV 25
#define SS 3
#define RR 8
#define BN_EPS 1e-5f

typedef float v2f __attribute__((ext_vector_type(2)));
typedef float v8f __attribute__((ext_vector_type(8)));

__device__ __forceinline__ v8f wmma_f32(v2f a, v2f b, v8f c) {
    // V_WMMA_F32_16X16X4_F32: D = A(16x4) * B(4x16) + C(16x16), all f32
    return __builtin_amdgcn_wmma_f32_16x16x4_f32(
        /*neg_a=*/false, a, /*neg_b=*/false, b,
        /*c_mod=*/(short)0, c, /*reuse_a=*/false, /*reuse_b=*/false);
}

// ---------------------------------------------------------------------------
// Kernel 1: xm[n,c,v] = mean over T of x[n,c,t,v]
// ---------------------------------------------------------------------------
__global__ void k_mean(const float* __restrict__ x, float* __restrict__ xm) {
    const int nc  = blockIdx.x;          // n*C + c
    const int tid = threadIdx.x;
    const int w = tid >> 5, l = tid & 31;
    __shared__ float sd[8 * 32];
    float p = 0.0f;
    if (l < VV) {
        const float* base = x + (size_t)nc * TT * VV;
        for (int t = w; t < TT; t += 8) p += base[t * VV + l];
    }
    sd[w * 32 + l] = p;
    __syncthreads();
    if (tid < VV) {
        float s = 0.0f;
        for (int ww = 0; ww < 8; ww++) s += sd[ww * 32 + tid];
        xm[nc * VV + tid] = s * (1.0f / (float)TT);
    }
}

// ---------------------------------------------------------------------------
// Kernel 2: attention path -> att padded to [N,S,C,32,32] (zeros in pad)
// att[n,s,o,u,v] = alpha*(sum_r w4[s,o,r]*tanh(x1[r,u]-x2[r,v]) + b4[s,o]) + A[s,u,v]
// ---------------------------------------------------------------------------
__global__ void k_att(const float* __restrict__ xm, const float* __restrict__ A,
                      const float* __restrict__ alpha,
                      const float* __restrict__ w1, const float* __restrict__ b1,
                      const float* __restrict__ w2, const float* __restrict__ b2,
                      const float* __restrict__ w4, const float* __restrict__ b4,
                      float* __restrict__ attp) {
    const int n = blockIdx.x, s = blockIdx.y, tid = threadIdx.x;
    __shared__ float sxm[CC * VV];
    __shared__ float sx1[RR * VV];
    __shared__ float sx2[RR * VV];
    __shared__ float sth[RR * VV * VV];

    for (int i = tid; i < CC * VV; i += 256) sxm[i] = xm[n * CC * VV + i];
    __syncthreads();

    if (tid < RR * VV) {
        const int r = tid / VV, v = tid - r * VV;
        float a1 = b1[s * RR + r];
        float a2 = b2[s * RR + r];
        const float* w1r = w1 + (s * RR + r) * CC;
        const float* w2r = w2 + (s * RR + r) * CC;
        for (int c = 0; c < CC; c++) {
            const float xv = sxm[c * VV + v];
            a1 += w1r[c] * xv;
            a2 += w2r[c] * xv;
        }
        sx1[tid] = a1;
        sx2[tid] = a2;
    }
    __syncthreads();

    for (int i = tid; i < RR * VV * VV; i += 256) {
        const int r = i / (VV * VV);
        const int uv = i - r * VV * VV;
        const int u = uv / VV, v = uv - u * VV;
        sth[i] = tanhf(sx1[r * VV + u] - sx2[r * VV + v]);
    }
    __syncthreads();

    const float al = alpha[0];
    float* outb = attp + (size_t)((n * SS + s) * CC) * 1024;
    for (int i = tid; i < CC * 1024; i += 256) {
        const int o = i >> 10;
        const int uvp = i & 1023;
        const int u = uvp >> 5, v = uvp & 31;
        float val = 0.0f;
        if (u < VV && v < VV) {
            float acc = b4[s * CC + o];
            const float* w4r = w4 + (s * CC + o) * RR;
            for (int r = 0; r < RR; r++)
                acc += w4r[r] * sth[(r * VV + u) * VV + v];
            val = acc * al + A[(s * VV + u) * VV + v];
        }
        outb[i] = val;
    }
}

// ---------------------------------------------------------------------------
// Kernel 3: fused double GEMM via WMMA f32 16x16x4, with inline BN reduction.
//   Stage x tile (s-invariant) into LDS once.
//   Phase 1 per s: X3[o16][t16][u pad32] = w3[s,o,:] . Xs + b3   (LDS)
//   Phase 2 per s: yacc[t,v] += X3[t,u] * att_pad[u,v]           (regs, over s)
//   Epilogue: store yraw + per-channel sum/sumsq atomics for BN.
// ---------------------------------------------------------------------------
__global__ void __launch_bounds__(256) k_gemm(const float* __restrict__ x,
                                              const float* __restrict__ w3,
                                              const float* __restrict__ b3,
                                              const float* __restrict__ attp,
                                              float* __restrict__ yraw,
                                              float* __restrict__ stats) {
    const int tb = blockIdx.x * 16;      // t tile base
    const int o0 = blockIdx.y * 16;      // o tile base
    const int n  = blockIdx.z;
    const int tid  = threadIdx.x;
    const int wave = tid >> 5;
    const int lane = tid & 31;
    const int hi = lane >> 4;            // lane group (0/1)
    const int lm = lane & 15;            // lane within group

    __shared__ float Xs[CC * 400];       // [c][col=(t_l*25+u)]  100 KB, s-invariant
    __shared__ float X3[16 * 16 * 32];   // [o_l][t_l][u pad 32]  32 KB
    __shared__ float sSum[16];
    __shared__ float sSq[16];

    // stage x tile once (reused for all 3 subsets s)
    {
        const float* xb = x + (size_t)(n * CC) * TT * VV;
        for (int i = tid; i < CC * 400; i += 256) {
            const int c = i / 400, col = i - c * 400;
            const int t = col / 25, u = col - t * 25;
            Xs[i] = xb[((size_t)c * TT + tb + t) * VV + u];
        }
    }
    for (int i = tid; i < 16 * 16 * 32; i += 256) X3[i] = 0.0f;  // zero u>=25 pad
    if (tid < 16) { sSum[tid] = 0.0f; sSq[tid] = 0.0f; }
    __syncthreads();

    v8f acc2[4];
    for (int j4 = 0; j4 < 4; j4++) acc2[j4] = (v8f){0, 0, 0, 0, 0, 0, 0, 0};

    for (int s = 0; s < SS; s++) {
        // ---------------- Phase 1: build X3 tile in LDS ----------------
        for (int nt = wave; nt < 25; nt += 8) {          // 25 N-tiles over 400 (t,u) cols
            const int col = nt * 16 + lm;                // this lane's column (B n-index)
            const int tl  = col / 25;
            const int u   = col - tl * 25;
            v8f acc = (v8f){0, 0, 0, 0, 0, 0, 0, 0};
            const float* w3b = w3 + (size_t)(s * CC + o0 + lm) * CC;  // A row m=lm
            const float* xsb = &Xs[col];                              // B col n
            for (int kk = 0; kk < 16; kk++) {            // K = C = 64, 4 per step
                const int k0 = kk * 4 + hi * 2;
                v2f a, b;
                a.x = w3b[k0 + 0];
                a.y = w3b[k0 + 1];
                b.x = xsb[(k0 + 0) * 400];
                b.y = xsb[(k0 + 1) * 400];
                acc = wmma_f32(a, b, acc);
            }
            for (int j = 0; j < 8; j++) {                // D: m = j + 8*hi, n = lm
                const int m = j + hi * 8;                // o_local
                X3[(m * 16 + tl) * 32 + u] = acc[j] + b3[s * CC + o0 + m];
            }
        }
        __syncthreads();

        // ---------------- Phase 2: y += X3 * att (K=u pad32, N=v pad32) --------
        for (int j4 = 0; j4 < 4; j4++) {
            const int job = wave * 4 + j4;               // 32 jobs = 16 o x 2 v-tiles
            const int ol = job >> 1, vt = job & 1;
            const float* ab = &X3[(ol * 16 + lm) * 32];  // A row m=lm (t_local)
            const float* bb = attp + ((size_t)((n * SS + s) * CC) + o0 + ol) * 1024
                              + vt * 16 + lm;            // B col n=lm (v)
            v8f acc = acc2[j4];
            for (int kk = 0; kk < 8; kk++) {             // K = 32 (u padded)
                const int u0 = kk * 4 + hi * 2;
                v2f a, b;
                a.x = ab[u0 + 0];
                a.y = ab[u0 + 1];
                b.x = bb[(u0 + 0) * 32];
                b.y = bb[(u0 + 1) * 32];
                acc = wmma_f32(a, b, acc);
            }
            acc2[j4] = acc;
        }
        __syncthreads();
    }

    // epilogue: store yraw[n,o,t,v] (v<25) and accumulate BN partials
    for (int j4 = 0; j4 < 4; j4++) {
        const int job = wave * 4 + j4;
        const int ol = job >> 1, vt = job & 1;
        const int v = vt * 16 + lm;
        if (v < VV) {
            float* yb = yraw + ((size_t)(n * CC + o0 + ol) * TT + tb) * VV + v;
            float ls = 0.0f, lq = 0.0f;
            for (int j = 0; j < 8; j++) {
                const int tl = j + hi * 8;
                const float val = acc2[j4][j];
                yb[tl * VV] = val;
                ls += val;
                lq += val * val;
            }
            atomicAdd(&sSum[ol], ls);
            atomicAdd(&sSq[ol], lq);
        }
    }
    __syncthreads();
    if (tid < 16) {
        atomicAdd(&stats[o0 + tid], sSum[tid]);
        atomicAdd(&stats[CC + o0 + tid], sSq[tid]);
    }
}

// ---------------------------------------------------------------------------
// Kernel 4: BN normalize + gamma/beta + residual + ReLU
// ---------------------------------------------------------------------------
__global__ void k_final(const float* __restrict__ yraw, const float* __restrict__ x,
                        const float* __restrict__ stats,
                        const float* __restrict__ gamma, const float* __restrict__ beta,
                        float* __restrict__ out) {
    const float invN = 1.0f / (float)(NN * TT * VV);
    const size_t total = (size_t)NN * CC * TT * VV;
    for (size_t idx = (size_t)blockIdx.x * 256 + threadIdx.x; idx < total;
         idx += (size_t)gridDim.x * 256) {
        const int o = (int)((idx / (TT * VV)) % CC);
        const float mean = stats[o] * invN;
        const float var  = stats[CC + o] * invN - mean * mean;
        float val = (yraw[idx] - mean) * rsqrtf(var + BN_EPS) * gamma[o] + beta[o] + x[idx];
        out[idx] = fmaxf(val, 0.0f);
    }
}

// ---------------------------------------------------------------------------
extern "C" void kernel_launch(void* const* d_in, const int* in_sizes, int n_in,
                              void* d_out, int out_size, void* d_ws, size_t ws_size,
                              hipStream_t stream) {
    (void)in_sizes; (void)n_in; (void)out_size; (void)ws_size;
    const float* x     = (const float*)d_in[0];
    const float* A     = (const float*)d_in[1];
    const float* alpha = (const float*)d_in[2];
    const float* w1    = (const float*)d_in[3];
    const float* b1    = (const float*)d_in[4];
    const float* w2    = (const float*)d_in[5];
    const float* b2    = (const float*)d_in[6];
    const float* w3    = (const float*)d_in[7];
    const float* b3    = (const float*)d_in[8];
    const float* w4    = (const float*)d_in[9];
    const float* b4    = (const float*)d_in[10];
    const float* gamma = (const float*)d_in[11];
    const float* beta  = (const float*)d_in[12];
    float* out = (float*)d_out;

    // workspace layout (floats)
    float* ws    = (float*)d_ws;
    float* xm    = ws;                                        // N*C*V = 102,400
    float* attp  = xm + (size_t)NN * CC * VV;                 // N*S*C*32*32 = 12,582,912
    float* yraw  = attp + (size_t)NN * SS * CC * 1024;        // N*C*T*V = 26,214,400
    float* stats = yraw + (size_t)NN * CC * TT * VV;          // 2*C

    hipMemsetAsync(stats, 0, 2 * CC * sizeof(float), stream);
    k_mean<<<NN * CC, 256, 0, stream>>>(x, xm);
    k_att<<<dim3(NN, SS), 256, 0, stream>>>(xm, A, alpha, w1, b1, w2, b2, w4, b4, attp);
    k_gemm<<<dim3(TT / 16, CC / 16, NN), 256, 0, stream>>>(x, w3, b3, attp, yraw, stats);
    k_final<<<4096, 256, 0, stream>>>(yraw, x, stats, gamma, beta, out);
}